// GNN_61692910240068
// MI455X (gfx1250) — compile-verified
//
#include <hip/hip_runtime.h>
#include <hip/hip_bf16.h>

typedef __attribute__((ext_vector_type(2))) float v2f;
typedef __attribute__((ext_vector_type(4))) float v4f;
typedef __attribute__((ext_vector_type(8))) float v8f;

#define F_IN  256   // K for both layers
#define F_HID 256
#define F_OUT 128
#define KSTEPS (F_IN / 4)   // 64 WMMA k-steps

// ---------------- degree / normalization ----------------
__global__ void k_init_deg(float* __restrict__ deg, int n) {
    int i = blockIdx.x * blockDim.x + threadIdx.x;
    if (i < n) deg[i] = 1.0f;               // self-loop contribution
}

__global__ void k_count_deg(const int* __restrict__ dst, float* __restrict__ deg, int e) {
    int i = blockIdx.x * blockDim.x + threadIdx.x;
    if (i < e) atomicAdd(&deg[dst[i]], 1.0f);
}

__global__ void k_dinv(float* __restrict__ d, int n) {
    int i = blockIdx.x * blockDim.x + threadIdx.x;
    if (i < n) d[i] = rsqrtf(d[i]);         // deg >= 1 always (self loops)
}

__global__ void k_norm(const int* __restrict__ src, const int* __restrict__ dst,
                       const float* __restrict__ dinv, float* __restrict__ nrm, int e) {
    int i = blockIdx.x * blockDim.x + threadIdx.x;
    if (i < e) nrm[i] = dinv[src[i]] * dinv[dst[i]];
}

// ---------------- weight repack ----------------
// Wp[((h*nout + c)*64 + t)*2 + {0,1}] = W[4t+2h + {0,1}][c]
// For fixed lane (h,c), consecutive k-steps are contiguous -> one b128 covers 2 steps.
__global__ void k_repack(const float* __restrict__ W, float* __restrict__ Wp, int nout) {
    int p = blockIdx.x * blockDim.x + threadIdx.x;
    int total = 2 * nout * KSTEPS;
    if (p >= total) return;
    int t = p & (KSTEPS - 1);               // k-step
    int hc = p >> 6;                        // h*nout + c
    int c = hc % nout;
    int h = hc / nout;
    int krow = 4 * t + 2 * h;
    Wp[(size_t)p * 2]     = W[(size_t)krow * nout + c];
    Wp[(size_t)p * 2 + 1] = W[(size_t)(krow + 1) * nout + c];
}

// ---------------- WMMA fp32 GEMM: H[M,NOUT] = act(X[M,256]) @ W[256,NOUT] ----------------
// One wave32 per 16-row tile; all NOUT/16 column tiles accumulated in registers.
// A 16x4 f32 frag: lane L holds row (L&15), VGPR v holds K = 2*(L>>4)+v.
// B 4x16 f32 frag: lane L col (L&15), VGPR v holds K-row = 2*(L>>4)+v.
// One v4f load from Wp supplies the B fragments of two consecutive k-steps.
// C/D 16x16 f32:   lane L col (L&15), VGPR v holds row v + 8*(L>>4).
template <int NOUT, bool RELU>
__global__ __launch_bounds__(32) void k_gemm_wmma(const float* __restrict__ X,
                                                  const float* __restrict__ Wp,
                                                  float* __restrict__ H, int nrows) {
    constexpr int NT = NOUT / 16;
    const int lane = threadIdx.x;
    const int half = lane >> 4;   // 0 or 1
    const int r    = lane & 15;
    const int row0 = blockIdx.x * 16;

    int rowA = row0 + r;
    if (rowA > nrows - 1) rowA = nrows - 1;   // clamp (keeps EXEC all-ones for WMMA)
    const float* xrow = X + (size_t)rowA * F_IN;
    const float* wlane = Wp + ((size_t)(half * NOUT + r) * KSTEPS) * 2;

    v8f acc[NT];
#pragma unroll
    for (int n = 0; n < NT; ++n)
#pragma unroll
        for (int j = 0; j < 8; ++j) acc[n][j] = 0.0f;

    for (int k = 0; k < F_IN; k += 8) {       // two WMMA k-steps per iteration
        v2f a0 = *(const v2f*)(xrow + k + 2 * half);
        v2f a1 = *(const v2f*)(xrow + k + 4 + 2 * half);
        if (RELU) {
            a0.x = fmaxf(a0.x, 0.0f); a0.y = fmaxf(a0.y, 0.0f);
            a1.x = fmaxf(a1.x, 0.0f); a1.y = fmaxf(a1.y, 0.0f);
        }
        const float* wp = wlane + (k >> 2) * 2;   // pair index t = k/4 (even)
        v4f b[NT];
#pragma unroll
        for (int n = 0; n < NT; ++n)
            b[n] = *(const v4f*)(wp + n * (16 * KSTEPS * 2));   // one b128 = 2 fragments
#pragma unroll
        for (int n = 0; n < NT; ++n) {
            v2f b0 = b[n].xy;
            acc[n] = __builtin_amdgcn_wmma_f32_16x16x4_f32(
                false, a0, false, b0, (short)0, acc[n], false, false);
        }
#pragma unroll
        for (int n = 0; n < NT; ++n) {
            v2f b1 = b[n].zw;
            acc[n] = __builtin_amdgcn_wmma_f32_16x16x4_f32(
                false, a1, false, b1, (short)0, acc[n], false, false);
        }
    }

#pragma unroll
    for (int n = 0; n < NT; ++n) {
#pragma unroll
        for (int v = 0; v < 8; ++v) {
            int rowD = row0 + v + 8 * half;
            if (rowD < nrows) H[(size_t)rowD * NOUT + n * 16 + r] = acc[n][v];
        }
    }
}

// ---------------- aggregation: out = dinv^2 * h (self loop) + bias ----------------
template <int F>
__global__ void k_agg_init(const float* __restrict__ H, const float* __restrict__ dinv,
                           const float* __restrict__ bias, float* __restrict__ out, int n) {
    long long idx = (long long)blockIdx.x * blockDim.x + threadIdx.x;
    if (idx >= (long long)n * F) return;
    int i = (int)(idx / F);
    int f = (int)(idx % F);
    float di = dinv[i];
    out[idx] = H[idx] * di * di + bias[f];
}

// ---------------- aggregation: out[dst] += norm * h[src] (float4 per thread) ----------------
// C = F/4 >= 32 chunks per edge => edge index is wave-uniform: force scalar so
// src/dst/norm become SMEM loads (3 s_load per wave instead of 96 lane loads).
template <int F>
__global__ void k_agg_edges(const float* __restrict__ H, const int* __restrict__ src,
                            const int* __restrict__ dst, const float* __restrict__ nrm,
                            float* __restrict__ out, int e) {
    constexpr int C = F / 4;                  // chunks per edge (64 or 32)
    long long idx = (long long)blockIdx.x * blockDim.x + threadIdx.x;
    if (idx >= (long long)e * C) return;
    int eg = (int)(idx >> (F == 256 ? 6 : 5));
    eg = __builtin_amdgcn_readfirstlane(eg);  // wave-uniform by construction
    int c = ((int)idx & (C - 1)) * 4;
    int s = src[eg], d = dst[eg];
    float nm = nrm[eg];
    float4 v = *(const float4*)(H + (size_t)s * F + c);
    float* op = out + (size_t)d * F + c;
    atomicAdd(op + 0, v.x * nm);
    atomicAdd(op + 1, v.y * nm);
    atomicAdd(op + 2, v.z * nm);
    atomicAdd(op + 3, v.w * nm);
}

// ---------------- launch ----------------
extern "C" void kernel_launch(void* const* d_in, const int* in_sizes, int n_in,
                              void* d_out, int out_size, void* d_ws, size_t ws_size,
                              hipStream_t stream) {
    const float* x  = (const float*)d_in[0];
    const int*   ei = (const int*)d_in[1];
    const float* W1 = (const float*)d_in[2];
    const float* b1 = (const float*)d_in[3];
    const float* W2 = (const float*)d_in[4];
    const float* b2 = (const float*)d_in[5];
    float* out = (float*)d_out;

    const int N = in_sizes[0] / F_IN;   // 50000
    const int E = in_sizes[1] / 2;      // 800000
    const int* src = ei;
    const int* dst = ei + E;

    char* ws = (char*)d_ws;
    size_t off = 0;
    auto alloc = [&](size_t bytes) -> void* {
        void* p = ws + off;
        off = (off + bytes + 255) & ~(size_t)255;
        return p;
    };
    float* dinv = (float*)alloc((size_t)N * 4);
    float* nrm  = (float*)alloc((size_t)E * 4);
    float* Wp1  = (float*)alloc((size_t)F_IN * F_HID * 4);
    float* Wp2  = (float*)alloc((size_t)F_IN * F_OUT * 4);
    float* h1   = (float*)alloc((size_t)N * F_HID * 4);  // reused as h2
    float* agg1 = (float*)alloc((size_t)N * F_HID * 4);
    float* h2 = h1;

    const int TB = 256;
    // normalization coefficients
    k_init_deg<<<(N + TB - 1) / TB, TB, 0, stream>>>(dinv, N);
    k_count_deg<<<(E + TB - 1) / TB, TB, 0, stream>>>(dst, dinv, E);
    k_dinv<<<(N + TB - 1) / TB, TB, 0, stream>>>(dinv, N);
    k_norm<<<(E + TB - 1) / TB, TB, 0, stream>>>(src, dst, dinv, nrm, E);

    // weight repacks (one-time, tiny)
    k_repack<<<(2 * F_HID * KSTEPS + TB - 1) / TB, TB, 0, stream>>>(W1, Wp1, F_HID);
    k_repack<<<(2 * F_OUT * KSTEPS + TB - 1) / TB, TB, 0, stream>>>(W2, Wp2, F_OUT);

    const int mt = (N + 15) / 16;       // 3125 row tiles

    // ----- layer 1 -----
    k_gemm_wmma<F_HID, false><<<mt, 32, 0, stream>>>(x, Wp1, h1, N);
    {
        long long w = (long long)N * F_HID;
        k_agg_init<F_HID><<<(unsigned)((w + TB - 1) / TB), TB, 0, stream>>>(h1, dinv, b1, agg1, N);
        long long we = (long long)E * (F_HID / 4);
        k_agg_edges<F_HID><<<(unsigned)((we + TB - 1) / TB), TB, 0, stream>>>(h1, src, dst, nrm, agg1, E);
    }

    // ----- layer 2 (ReLU folded into A-fragment loads) -----
    k_gemm_wmma<F_OUT, true><<<mt, 32, 0, stream>>>(agg1, Wp2, h2, N);
    {
        long long w = (long long)N * F_OUT;
        k_agg_init<F_OUT><<<(unsigned)((w + TB - 1) / TB), TB, 0, stream>>>(h2, dinv, b2, out, N);
        long long we = (long long)E * (F_OUT / 4);
        k_agg_edges<F_OUT><<<(unsigned)((we + TB - 1) / TB), TB, 0, stream>>>(h2, src, dst, nrm, out, E);
    }
}